// GATConv_82394652606637
// MI455X (gfx1250) — compile-verified
//
#include <hip/hip_runtime.h>
#include <hip/hip_bf16.h>
#include <cmath>

#define BATCH 64
#define NN    512
#define INF_  128
#define OUTF  128

typedef __attribute__((ext_vector_type(2))) float v2f;
typedef __attribute__((ext_vector_type(8))) float v8f;
typedef __attribute__((ext_vector_type(4))) unsigned int u32x4;
typedef __attribute__((ext_vector_type(8))) int          i32x8;
typedef __attribute__((ext_vector_type(4))) int          i32x4;

#if __has_builtin(__builtin_amdgcn_tensor_load_to_lds) && \
    __has_builtin(__builtin_amdgcn_s_wait_tensorcnt)
#define USE_TDM 1
#else
#define USE_TDM 0
#endif

// Padded LDS row stride for staged h chunks: 128 data dwords + 16 pad dwords.
// B-fragment ds_load: lane-half0 reads row k, half1 reads row k+2 ->
// address delta = 2*144 = 288 dwords == 32 (mod 64 banks) -> disjoint banks.
#define HSTRIDE 144

// ---------------------------------------------------------------------------
// Kernel 1: h = x @ W  (f32 WMMA 16x16x4), fused src/dst = h @ a_src|a_dst
// via ds_add_f32 reductions. grid = B*(N/16), 256 threads = 8 waves.
// ---------------------------------------------------------------------------
__global__ __launch_bounds__(256) void gat_h_kernel(
    const float* __restrict__ x, const float* __restrict__ W,
    const float* __restrict__ a_src, const float* __restrict__ a_dst,
    float* __restrict__ h, float* __restrict__ src, float* __restrict__ dst) {
  __shared__ float sh_s[16];
  __shared__ float sh_d[16];

  const int tile = blockIdx.x;
  const int b    = tile >> 5;
  const int i0   = (tile & 31) << 4;
  const int t    = threadIdx.x;
  const int wave = t >> 5;
  const int lane = t & 31;
  const int ln   = lane & 15;           // A: row m; B/C/D: col n
  const int hh   = lane >> 4;           // lane half
  const int c0   = wave << 4;

  if (t < 16) { sh_s[t] = 0.f; sh_d[t] = 0.f; }
  __syncthreads();

  const float* xrow = x + ((size_t)b * NN + (i0 + ln)) * (size_t)INF_;

  v8f acc = {};
  #pragma unroll 4
  for (int k = 0; k < INF_; k += 4) {
    v2f a, bm;
    a.x  = xrow[k + 2 * hh];
    a.y  = xrow[k + 2 * hh + 1];
    bm.x = W[(size_t)(k + 2 * hh) * OUTF + c0 + ln];
    bm.y = W[(size_t)(k + 2 * hh + 1) * OUTF + c0 + ln];
    acc = __builtin_amdgcn_wmma_f32_16x16x4_f32(false, a, false, bm,
                                                (short)0, acc, false, false);
  }

  // store h tile + accumulate per-row attention projections
  float* hb = h + (size_t)b * NN * OUTF;
  const float as = a_src[c0 + ln];
  const float ad = a_dst[c0 + ln];
  #pragma unroll
  for (int v = 0; v < 8; ++v) {
    const int row = 8 * hh + v;
    hb[(size_t)(i0 + row) * OUTF + c0 + ln] = acc[v];
    atomicAdd(&sh_s[row], acc[v] * as);   // ds_add_f32
    atomicAdd(&sh_d[row], acc[v] * ad);
  }
  __syncthreads();
  if (t < 16) {
    src[(size_t)b * NN + i0 + t] = sh_s[t];
    dst[(size_t)b * NN + i0 + t] = sh_d[t];
  }
}

// ---------------------------------------------------------------------------
// TDM issue: 1D copy of 2048 f32 (16 rows x 128) h[b][j0..j0+15][:] -> LDS,
// with LDS padding of 16 dwords every 128 dwords (row stride HSTRIDE=144).
// Descriptor packing per CDNA5 ISA §8.3/8.4 (D# groups 0/1; groups 2/3 null).
// Uses the 6-arg builtin form (clang-23 / therock-10.0 headers).
// ---------------------------------------------------------------------------
#if USE_TDM
__device__ __forceinline__ void tdm_load_chunk(const float* gsrc,
                                               unsigned lds_byte_addr) {
  const unsigned long long ga = (unsigned long long)(uintptr_t)gsrc;
  u32x4 g0;
  g0[0] = 1u;                                           // count=1 (valid), user mode
  g0[1] = lds_byte_addr;                                // lds_addr  [63:32]
  g0[2] = (unsigned)(ga & 0xffffffffu);                 // global_addr[31:0]
  g0[3] = (unsigned)((ga >> 32) & 0x1ffffffu) | (2u << 30); // ga[56:32] | type=2

  i32x8 g1;
  g1[0] = (2 << 16)      // data_size = 4B
        | (1 << 20)      // pad_enable
        | (6 << 22)      // pad_interval: 128 dwords
        | (15 << 25);    // pad_amount: 16 dwords
  g1[1] = (2048 & 0xffff) << 16;          // tensor_dim0[15:0]  (bits 63:48)
  g1[2] = (2048 >> 16) | (1 << 16);       // tensor_dim0[31:16] | tensor_dim1 lo16=1
  g1[3] = (2048 << 16);                   // tensor_dim1 hi16=0 | tile_dim0=2048
  g1[4] = 1;                              // tile_dim1=1, tile_dim2=0
  g1[5] = 2048;                           // tensor_dim0_stride lo32
  g1[6] = (2048 & 0xffff) << 16;          // dim0_stride hi16=0 | dim1_stride lo16
  g1[7] = 0;                              // dim1_stride hi32
  i32x4 gz4 = {0, 0, 0, 0};
  i32x8 gz8 = {0, 0, 0, 0, 0, 0, 0, 0};
  __builtin_amdgcn_tensor_load_to_lds(g0, g1, gz4, gz4, gz8, 0);
}
#endif

// ---------------------------------------------------------------------------
// Kernel 2: masked-softmax attention + (alpha @ h) + ELU.
// grid = B*(N/16), 256 threads = 8 waves.
// Pass 1 computes logits e once (adj read ONCE), caches them in LDS, and
// builds per-row online (max, sumexp). Pass 2 streams 16-wide j chunks:
// alpha tile from cached logits; h chunk staged in LDS by the Tensor Data
// Mover (double-buffered, prefetch of chunk 0 overlaps all of pass 1);
// 8 waves accumulate 16x16 output tiles with f32 WMMA. ELU fused in store.
// ---------------------------------------------------------------------------
__global__ __launch_bounds__(256) void gat_attn_kernel(
    const int* __restrict__ adj, const float* __restrict__ h,
    const float* __restrict__ src, const float* __restrict__ dst,
    float* __restrict__ out) {
  __shared__ float sh_e[16][528];          // cached logits, 528 = 16 mod 64 pad
  __shared__ float sh_h[2][16 * HSTRIDE];  // TDM-staged h chunks (padded rows)
  __shared__ float sh_alpha[16][17];
  __shared__ float red_m[16][16];
  __shared__ float red_l[16][16];
  __shared__ float sh_m[16];
  __shared__ float sh_l[16];

  const int tile = blockIdx.x;
  const int b    = tile >> 5;
  const int i0   = (tile & 31) << 4;
  const int t    = threadIdx.x;
  const int wave = t >> 5;
  const int lane = t & 31;
  const int r    = t >> 4;             // attention row 0..15
  const int jj   = t & 15;             // sub-column 0..15
  const int ln   = lane & 15;
  const int hh   = lane >> 4;
  const int c0   = wave << 4;

  const float* hbase = h + (size_t)b * NN * OUTF;

#if USE_TDM
  // prefetch chunk 0 immediately; DMA overlaps pass 1 entirely
  if (wave == 0)
    tdm_load_chunk(hbase, (unsigned)(uintptr_t)&sh_h[0][0]);
#endif

  const int*   adjrow = adj + ((size_t)b * NN + (i0 + r)) * (size_t)NN;
  const float  src_i  = src[(size_t)b * NN + i0 + r];
  const float* dstb   = dst + (size_t)b * NN;

  // ---- pass 1: compute + cache masked logits, online (max, sumexp)
  float m = -1e30f, l = 0.f;
  for (int j = jj; j < NN; j += 16) {
    float e = src_i + dstb[j];
    e = (e > 0.f) ? e : 0.2f * e;            // leaky_relu(0.2)
    e = adjrow[j] ? e : -1e9f;               // mask (adj read once)
    sh_e[r][j] = e;
    if (e > m) { l *= __expf(m - e); m = e; }
    l += __expf(e - m);
  }
  red_m[r][jj] = m;
  red_l[r][jj] = l;
  __syncthreads();
  if (jj == 0) {
    float M = red_m[r][0], L = red_l[r][0];
    #pragma unroll
    for (int q = 1; q < 16; ++q) {
      float mq = red_m[r][q], lq = red_l[r][q];
      float Mn = fmaxf(M, mq);
      L = L * __expf(M - Mn) + lq * __expf(mq - Mn);
      M = Mn;
    }
    sh_m[r] = M;
    sh_l[r] = L;
  }
  __syncthreads();

  const float row_m   = sh_m[r];
  const float row_inv = 1.0f / sh_l[r];

  // ---- pass 2: 32 chunks of 16 j's, double-buffered TDM + WMMA
  v8f acc = {};
  for (int c = 0; c < NN / 16; ++c) {
    const int j0  = c << 4;
    const int buf = c & 1;

#if !USE_TDM
    // fallback: cooperative staged copy into padded layout
    for (int e = t; e < 2048; e += 256) {
      sh_h[buf][(e >> 7) * HSTRIDE + (e & 127)] = hbase[(size_t)j0 * OUTF + e];
    }
#endif

    // alpha tile from cached logits (exact masking: stored -1e9 marks adj==0)
    {
      const float e = sh_e[r][j0 + jj];
      sh_alpha[r][jj] = (e == -1e9f) ? 0.f : __expf(e - row_m) * row_inv;
    }
#if USE_TDM
    if (wave == 0) __builtin_amdgcn_s_wait_tensorcnt(0);   // chunk c landed
#endif
    __syncthreads();
#if USE_TDM
    if (wave == 0 && c + 1 < NN / 16)                      // prefetch chunk c+1
      tdm_load_chunk(hbase + (size_t)(j0 + 16) * OUTF,
                     (unsigned)(uintptr_t)&sh_h[buf ^ 1][0]);
#endif

    const float* hs = &sh_h[buf][0];
    #pragma unroll
    for (int k = 0; k < 16; k += 4) {
      v2f a, bm;
      a.x  = sh_alpha[ln][k + 2 * hh];
      a.y  = sh_alpha[ln][k + 2 * hh + 1];
      const float* hrow = hs + (k + 2 * hh) * HSTRIDE + c0 + ln;
      bm.x = hrow[0];
      bm.y = hrow[HSTRIDE];
      acc = __builtin_amdgcn_wmma_f32_16x16x4_f32(false, a, false, bm,
                                                  (short)0, acc, false, false);
    }
    __syncthreads();   // protect sh_alpha (+ fallback sh_h) for next chunk
  }

  // ---- fused ELU + store
  float* ob = out + (size_t)b * NN * OUTF;
  #pragma unroll
  for (int v = 0; v < 8; ++v) {
    float xv = acc[v];
    xv = (xv > 0.f) ? xv : (__expf(xv) - 1.f);
    ob[(size_t)(i0 + 8 * hh + v) * OUTF + c0 + ln] = xv;
  }
}

// ---------------------------------------------------------------------------
extern "C" void kernel_launch(void* const* d_in, const int* in_sizes, int n_in,
                              void* d_out, int out_size, void* d_ws, size_t ws_size,
                              hipStream_t stream) {
  (void)in_sizes; (void)n_in; (void)out_size; (void)ws_size;
  const int*   adj   = (const int*)d_in[0];
  const float* x     = (const float*)d_in[1];
  const float* W     = (const float*)d_in[2];
  const float* a_src = (const float*)d_in[3];
  const float* a_dst = (const float*)d_in[4];
  float* out = (float*)d_out;

  // workspace: h [B*N*OUTF] | src [B*N] | dst [B*N]   (~17 MB)
  float* h   = (float*)d_ws;
  float* src = h + (size_t)BATCH * NN * OUTF;
  float* dst = src + (size_t)BATCH * NN;

  gat_h_kernel<<<BATCH * (NN / 16), 256, 0, stream>>>(x, W, a_src, a_dst,
                                                      h, src, dst);
  gat_attn_kernel<<<BATCH * (NN / 16), 256, 0, stream>>>(adj, h, src, dst, out);
}